// CGIteration_12532714570336
// MI455X (gfx1250) — compile-verified
//
#include <hip/hip_runtime.h>

// CDNA5 (gfx1250) CG tensor-product kernel.
// Store-bandwidth-bound problem (~639MB fp32 out @ 23.3TB/s => ~27us floor).
// Each 16x16 (p,q) tile computed with chained V_WMMA_F32_16X16X4_F32:
//   A[p,j] = sum_i x1[i,p]*C[i,j,M]   (VALU, K<=7)
//   B[j,q] = x2[j,q]                  (direct from registers)
//   D = A*B via 1 or 2 K=4 fp32 WMMAs, stored with non-temporal b32 stores.

typedef __attribute__((ext_vector_type(2))) float v2f;
typedef __attribute__((ext_vector_type(8))) float v8f;

#define DEVINL __device__ __forceinline__

DEVINL v8f wmma_f32_k4(v2f a, v2f b, v8f c) {
  // 8 args: (neg_a, A, neg_b, B, c_mod, C, reuse_a, reuse_b)
  return __builtin_amdgcn_wmma_f32_16x16x4_f32(false, a, false, b, (short)0, c,
                                               false, false);
}

// One combo (l1,l2,L): computes (2L+1) 16x16 tiles for one sample.
// BASE = output base of this combo's (L,S)-key, NB = #blocks in key,
// BLK = this combo's block index within the key.
template <int L1, int L2, int LL, int BASE, int NB, int BLK>
DEVINL void do_combo(const float (&x1r)[16], const float (&x2r)[16],
                     const float* __restrict__ cg, float* __restrict__ outn,
                     int col, bool hi) {
  constexpr int O1 = (L1 == 0) ? 0 : (L1 == 1) ? 1 : (L1 == 2) ? 4 : 9;
  constexpr int O2 = (L2 == 0) ? 0 : (L2 == 1) ? 1 : (L2 == 2) ? 4 : 9;
  constexpr int I = 2 * L1 + 1;           // contraction over i (VALU)
  constexpr int J = 2 * L2 + 1;           // WMMA K dimension (<=7)
  constexpr int NC = (J + 3) / 4;         // number of K=4 WMMA chunks
  constexpr int CB = L1 * 5488 + L2 * 1372 + LL * 343;  // cg[l1,l2,L,...]

  // B operand (x2 rows) per chunk: VGPR0 holds rows {4c,4c+2}, VGPR1 {4c+1,4c+3}
  // split across lane halves; invalid rows zeroed so padded K contributes 0.
  v2f b[NC];
#pragma unroll
  for (int c = 0; c < NC; ++c) {
    float lox = (4 * c + 0 < J) ? x2r[O2 + 4 * c + 0] : 0.0f;
    float loy = (4 * c + 1 < J) ? x2r[O2 + 4 * c + 1] : 0.0f;
    float hix = (4 * c + 2 < J) ? x2r[O2 + 4 * c + 2] : 0.0f;
    float hiy = (4 * c + 3 < J) ? x2r[O2 + 4 * c + 3] : 0.0f;
    b[c].x = hi ? hix : lox;
    b[c].y = hi ? hiy : loy;
  }

#pragma unroll 1  // keep M rolled: bounds code size, cg reads stay scalar loads
  for (int M = 0; M < 2 * LL + 1; ++M) {
    v8f acc = {};
#pragma unroll
    for (int c = 0; c < NC; ++c) {
      // A[p, j] = sum_i x1[i,p] * C[i,j,M] for the 4 j's of this chunk.
      float s[4];
#pragma unroll
      for (int t = 0; t < 4; ++t) {
        int j = 4 * c + t;
        float acc_s = 0.0f;
        if (j < J) {  // compile-time after unroll
#pragma unroll
          for (int i = 0; i < I; ++i)
            acc_s = fmaf(x1r[O1 + i], cg[CB + i * 49 + j * 7 + M], acc_s);
        }
        s[t] = acc_s;
      }
      v2f a;
      a.x = hi ? s[2] : s[0];  // lane<16: K=4c,4c+1 ; lane>=16: K=4c+2,4c+3
      a.y = hi ? s[3] : s[1];
      acc = wmma_f32_k4(a, b[c], acc);
    }
    // D layout: VGPR r = row r (lanes 0-15) / row r+8 (lanes 16-31), col = lane&15.
    float* tout = outn + BASE + BLK * 256 + M * (256 * NB);
    const int r0 = hi ? 8 : 0;
#pragma unroll
    for (int r = 0; r < 8; ++r)
      __builtin_nontemporal_store(acc[r], tout + (r0 + r) * 16 + col);
  }
}

__global__ __launch_bounds__(256) void cg_tensor_product_kernel(
    const float* __restrict__ x1_0, const float* __restrict__ x2_0,
    const float* __restrict__ x1_1, const float* __restrict__ x2_1,
    const float* __restrict__ x1_2, const float* __restrict__ x2_2,
    const float* __restrict__ x1_3, const float* __restrict__ x2_3,
    const float* __restrict__ cg, float* __restrict__ out, int nsamp) {
  const int wave = threadIdx.x >> 5;
  const int lane = threadIdx.x & 31;
  const int n = blockIdx.x * (blockDim.x >> 5) + wave;
  if (n >= nsamp) return;  // wave-uniform: EXEC stays all-1s for WMMA

  const int col = lane & 15;
  const bool hi = lane >= 16;

  // Preload this sample's x1/x2 lane-columns: x[l][n, i, col], packed at
  // offsets {l0:0, l1:1..3, l2:4..8, l3:9..15}.
  float x1r[16], x2r[16];
  x1r[0] = x1_0[n * 16 + col];
  x2r[0] = x2_0[n * 16 + col];
#pragma unroll
  for (int i = 0; i < 3; ++i) {
    x1r[1 + i] = x1_1[(n * 3 + i) * 16 + col];
    x2r[1 + i] = x2_1[(n * 3 + i) * 16 + col];
  }
#pragma unroll
  for (int i = 0; i < 5; ++i) {
    x1r[4 + i] = x1_2[(n * 5 + i) * 16 + col];
    x2r[4 + i] = x2_2[(n * 5 + i) * 16 + col];
  }
#pragma unroll
  for (int i = 0; i < 7; ++i) {
    x1r[9 + i] = x1_3[(n * 7 + i) * 16 + col];
    x2r[9 + i] = x2_3[(n * 7 + i) * 16 + col];
  }

  float* outn = out + (size_t)n * 39936;

  // 34 combos in reference order; (BASE, NB, BLK) from the (L,S)-sorted layout.
  do_combo<0, 0, 0, 0, 4, 0>(x1r, x2r, cg, outn, col, hi);
  do_combo<0, 1, 1, 3328, 6, 0>(x1r, x2r, cg, outn, col, hi);
  do_combo<0, 2, 2, 13056, 7, 0>(x1r, x2r, cg, outn, col, hi);
  do_combo<0, 3, 3, 29184, 6, 0>(x1r, x2r, cg, outn, col, hi);
  do_combo<1, 0, 1, 3328, 6, 1>(x1r, x2r, cg, outn, col, hi);
  do_combo<1, 1, 0, 0, 4, 1>(x1r, x2r, cg, outn, col, hi);
  do_combo<1, 1, 1, 1024, 3, 0>(x1r, x2r, cg, outn, col, hi);
  do_combo<1, 1, 2, 13056, 7, 1>(x1r, x2r, cg, outn, col, hi);
  do_combo<1, 2, 1, 3328, 6, 2>(x1r, x2r, cg, outn, col, hi);
  do_combo<1, 2, 2, 7936, 4, 0>(x1r, x2r, cg, outn, col, hi);
  do_combo<1, 2, 3, 29184, 6, 1>(x1r, x2r, cg, outn, col, hi);
  do_combo<1, 3, 2, 13056, 7, 2>(x1r, x2r, cg, outn, col, hi);
  do_combo<1, 3, 3, 22016, 4, 0>(x1r, x2r, cg, outn, col, hi);
  do_combo<2, 0, 2, 13056, 7, 3>(x1r, x2r, cg, outn, col, hi);
  do_combo<2, 1, 1, 3328, 6, 3>(x1r, x2r, cg, outn, col, hi);
  do_combo<2, 1, 2, 7936, 4, 1>(x1r, x2r, cg, outn, col, hi);
  do_combo<2, 1, 3, 29184, 6, 2>(x1r, x2r, cg, outn, col, hi);
  do_combo<2, 2, 0, 0, 4, 2>(x1r, x2r, cg, outn, col, hi);
  do_combo<2, 2, 1, 1024, 3, 1>(x1r, x2r, cg, outn, col, hi);
  do_combo<2, 2, 2, 13056, 7, 4>(x1r, x2r, cg, outn, col, hi);
  do_combo<2, 2, 3, 22016, 4, 1>(x1r, x2r, cg, outn, col, hi);
  do_combo<2, 3, 1, 3328, 6, 4>(x1r, x2r, cg, outn, col, hi);
  do_combo<2, 3, 2, 7936, 4, 2>(x1r, x2r, cg, outn, col, hi);
  do_combo<2, 3, 3, 29184, 6, 3>(x1r, x2r, cg, outn, col, hi);
  do_combo<3, 0, 3, 29184, 6, 4>(x1r, x2r, cg, outn, col, hi);
  do_combo<3, 1, 2, 13056, 7, 5>(x1r, x2r, cg, outn, col, hi);
  do_combo<3, 1, 3, 22016, 4, 2>(x1r, x2r, cg, outn, col, hi);
  do_combo<3, 2, 1, 3328, 6, 5>(x1r, x2r, cg, outn, col, hi);
  do_combo<3, 2, 2, 7936, 4, 3>(x1r, x2r, cg, outn, col, hi);
  do_combo<3, 2, 3, 29184, 6, 5>(x1r, x2r, cg, outn, col, hi);
  do_combo<3, 3, 0, 0, 4, 3>(x1r, x2r, cg, outn, col, hi);
  do_combo<3, 3, 1, 1024, 3, 2>(x1r, x2r, cg, outn, col, hi);
  do_combo<3, 3, 2, 13056, 7, 6>(x1r, x2r, cg, outn, col, hi);
  do_combo<3, 3, 3, 22016, 4, 3>(x1r, x2r, cg, outn, col, hi);
}

extern "C" void kernel_launch(void* const* d_in, const int* in_sizes, int n_in,
                              void* d_out, int out_size, void* d_ws,
                              size_t ws_size, hipStream_t stream) {
  // setup_inputs() dict order: x1_l0, x2_l0, x1_l1, x2_l1, ..., x1_l3, x2_l3, cg
  const float* x1_0 = (const float*)d_in[0];
  const float* x2_0 = (const float*)d_in[1];
  const float* x1_1 = (const float*)d_in[2];
  const float* x2_1 = (const float*)d_in[3];
  const float* x1_2 = (const float*)d_in[4];
  const float* x2_2 = (const float*)d_in[5];
  const float* x1_3 = (const float*)d_in[6];
  const float* x2_3 = (const float*)d_in[7];
  const float* cg = (const float*)d_in[8];
  float* out = (float*)d_out;

  const int nsamp = in_sizes[0] / 16;  // x1_l0 is (N, 1, 16)
  const int wavesPerBlock = 8;         // 256 threads = 8 wave32
  const int blocks = (nsamp + wavesPerBlock - 1) / wavesPerBlock;

  cg_tensor_product_kernel<<<blocks, 32 * wavesPerBlock, 0, stream>>>(
      x1_0, x2_0, x1_1, x2_1, x1_2, x2_2, x1_3, x2_3, cg, out, nsamp);
}